// BahdanauAttention_68891275428028
// MI455X (gfx1250) — compile-verified
//
#include <hip/hip_runtime.h>

// Problem constants (match reference)
#define B_ 4
#define Q_ 256
#define V_ 2048
#define H_ 512
#define U_ 128
#define KT_ 64   // K-chunk staged in LDS per double-buffer slot (kernel 3)

typedef __attribute__((ext_vector_type(2))) float v2f;
typedef __attribute__((ext_vector_type(8))) float v8f;
typedef __attribute__((ext_vector_type(4))) int v4i;

typedef __attribute__((address_space(1))) v4i as1_v4i;  // global int4
typedef __attribute__((address_space(3))) v4i as3_v4i;  // LDS int4

__device__ __forceinline__ float dtanh(float x) {
#if __has_builtin(__builtin_amdgcn_tanhf)
  return __builtin_amdgcn_tanhf(x);   // native V_TANH_F32 on gfx1250
#else
  return tanhf(x);
#endif
}

// ---- CDNA5 async global -> LDS copy (16 bytes / lane), ASYNCcnt-tracked ----
__device__ __forceinline__ void async_copy16(const float* g, float* l) {
#if __has_builtin(__builtin_amdgcn_global_load_async_to_lds_b128)
  as1_v4i* gp = (as1_v4i*)(unsigned long long)(const void*)g;
  as3_v4i* lp = (as3_v4i*)(unsigned)(unsigned long long)(void*)l;  // LDS offset = flat low 32b
  __builtin_amdgcn_global_load_async_to_lds_b128(gp, lp, 0, 0);
#else
  unsigned long long ga = (unsigned long long)(const void*)g;
  unsigned lo = (unsigned)(unsigned long long)(void*)l;
  asm volatile("global_load_async_to_lds_b128 %0, %1, off"
               :: "v"(lo), "v"(ga) : "memory");
#endif
}

__device__ __forceinline__ void wait_async2() {
#if __has_builtin(__builtin_amdgcn_s_wait_asynccnt)
  __builtin_amdgcn_s_wait_asynccnt(2);
#else
  asm volatile("s_wait_asynccnt 0x2" ::: "memory");
#endif
}
__device__ __forceinline__ void wait_async0() {
#if __has_builtin(__builtin_amdgcn_s_wait_asynccnt)
  __builtin_amdgcn_s_wait_asynccnt(0);
#else
  asm volatile("s_wait_asynccnt 0x0" ::: "memory");
#endif
}

// ---------------------------------------------------------------------------
// Kernel 1: fused projection GEMM.
// Rows [0,1024)   : pq = queries @ w1   (queries is [B*Q, H] contiguous)
// Rows [1024,9216): pv = values  @ w2   (values  is [B*V, H] contiguous)
// One wave computes a 64x16 fp32 tile via 4x V_WMMA_F32_16X16X4_F32 per K-step.
// ---------------------------------------------------------------------------
__global__ __launch_bounds__(128) void proj_kernel(
    const float* __restrict__ queries, const float* __restrict__ values,
    const float* __restrict__ w1, const float* __restrict__ w2,
    float* __restrict__ pq, float* __restrict__ pv) {
  const int wave = blockIdx.x * (blockDim.x >> 5) + (threadIdx.x >> 5);
  const int lane = threadIdx.x & 31;
  const int NT = U_ / 16;                 // 8 column tiles
  const int mb = wave / NT;               // 0..143 (64-row blocks)
  const int nt = wave % NT;
  const int m0 = mb * 64;
  const int n0 = nt * 16;

  const float* src; const float* w; float* dst; int row0;
  if (m0 < B_ * Q_) { src = queries; w = w1; dst = pq; row0 = m0; }
  else              { src = values;  w = w2; dst = pv; row0 = m0 - B_ * Q_; }

  const int mi = lane & 15;               // A: lane -> M row
  const int kk = (lane >> 4) << 1;        // A/B: upper half-wave -> K+2
  const float* a0 = src + (size_t)(row0 + mi) * H_;
  const float* wc = w + n0 + (lane & 15); // B: lane -> N col

  v8f acc0 = {}, acc1 = {}, acc2 = {}, acc3 = {};
  for (int k0 = 0; k0 < H_; k0 += 4) {
    v2f b;
    b.x = wc[(size_t)(k0 + kk) * U_];
    b.y = wc[(size_t)(k0 + kk + 1) * U_];
    v2f a_0 = *(const v2f*)(a0 + 0 * 16 * H_ + k0 + kk);
    v2f a_1 = *(const v2f*)(a0 + 1 * 16 * H_ + k0 + kk);
    v2f a_2 = *(const v2f*)(a0 + 2 * 16 * H_ + k0 + kk);
    v2f a_3 = *(const v2f*)(a0 + 3 * 16 * H_ + k0 + kk);
    acc0 = __builtin_amdgcn_wmma_f32_16x16x4_f32(false, a_0, false, b, (short)0, acc0, false, false);
    acc1 = __builtin_amdgcn_wmma_f32_16x16x4_f32(false, a_1, false, b, (short)0, acc1, false, false);
    acc2 = __builtin_amdgcn_wmma_f32_16x16x4_f32(false, a_2, false, b, (short)0, acc2, false, false);
    acc3 = __builtin_amdgcn_wmma_f32_16x16x4_f32(false, a_3, false, b, (short)0, acc3, false, false);
  }
  const int mo = (lane >> 4) * 8;         // C/D: upper half-wave -> M+8
  const int n  = n0 + (lane & 15);
  for (int r = 0; r < 8; ++r) {
    dst[(size_t)(row0 + 0 * 16 + mo + r) * U_ + n] = acc0[r];
    dst[(size_t)(row0 + 1 * 16 + mo + r) * U_ + n] = acc1[r];
    dst[(size_t)(row0 + 2 * 16 + mo + r) * U_ + n] = acc2[r];
    dst[(size_t)(row0 + 3 * 16 + mo + r) * U_ + n] = acc3[r];
  }
}

// ---------------------------------------------------------------------------
// Kernel 2: fused scores + softmax. One workgroup (8 waves) per (b,q).
// Never materializes the [B,Q,V,U] tensor: lane holds 4 u-slots of pq and v,
// streams pv rows (float4, fully coalesced), tanh+dot in registers,
// wave32 shfl_xor reduction, LDS softmax over V=2048.
// ---------------------------------------------------------------------------
__global__ __launch_bounds__(256) void score_softmax_kernel(
    const float* __restrict__ pq, const float* __restrict__ pv,
    const float* __restrict__ vvec, float* __restrict__ attn) {
  __shared__ float s_scores[V_];
  __shared__ float s_tmp[256];
  __shared__ float s_pq[U_];
  __shared__ float s_v[U_];

  const int tid  = threadIdx.x;
  const int lane = tid & 31;
  const int wave = tid >> 5;
  const int bq = blockIdx.x;              // b*Q + q
  const int b  = bq / Q_;

  if (tid < U_)            s_pq[tid] = pq[(size_t)bq * U_ + tid];
  else if (tid < 2 * U_)   s_v[tid - U_] = vvec[tid - U_];
  __syncthreads();

  float4 q4 = *(const float4*)(s_pq + lane * 4);
  float4 v4 = *(const float4*)(s_v + lane * 4);

  const float* pvb = pv + (size_t)b * V_ * U_;
  for (int i = 0; i < V_ / 8; ++i) {
    const int vr = wave + 8 * i;
    float4 p4 = *(const float4*)(pvb + (size_t)vr * U_ + lane * 4);
    float t = v4.x * dtanh(q4.x + p4.x)
            + v4.y * dtanh(q4.y + p4.y)
            + v4.z * dtanh(q4.z + p4.z)
            + v4.w * dtanh(q4.w + p4.w);
    for (int off = 16; off > 0; off >>= 1) t += __shfl_xor(t, off, 32);
    if (lane == 0) s_scores[vr] = t;
  }
  __syncthreads();

  float lmax = -3.402823466e38f;
  for (int j = 0; j < V_ / 256; ++j) lmax = fmaxf(lmax, s_scores[tid + 256 * j]);
  s_tmp[tid] = lmax; __syncthreads();
  for (int s = 128; s > 0; s >>= 1) {
    if (tid < s) s_tmp[tid] = fmaxf(s_tmp[tid], s_tmp[tid + s]);
    __syncthreads();
  }
  const float gmax = s_tmp[0];
  __syncthreads();

  float lsum = 0.f;
  for (int j = 0; j < V_ / 256; ++j) {
    float e = __expf(s_scores[tid + 256 * j] - gmax);
    s_scores[tid + 256 * j] = e;
    lsum += e;
  }
  s_tmp[tid] = lsum; __syncthreads();
  for (int s = 128; s > 0; s >>= 1) {
    if (tid < s) s_tmp[tid] += s_tmp[tid + s];
    __syncthreads();
  }
  const float inv = 1.0f / s_tmp[0];

  for (int j = 0; j < V_ / 256; ++j)
    attn[(size_t)bq * V_ + tid + 256 * j] = s_scores[tid + 256 * j] * inv;
}

// ---------------------------------------------------------------------------
// Kernel 3: out[b] = attn[b] (QxV) @ values[b] (VxH), fp32 WMMA.
// One block (4 waves) owns a full-Q x 16-column strip: the 64x16 B tile for
// each K-chunk is staged once into LDS with async global->LDS B128 copies
// (double-buffered, ASYNCcnt-synchronized) and shared by all 4 waves.
// ---------------------------------------------------------------------------
__global__ __launch_bounds__(128) void out_gemm_kernel(
    const float* __restrict__ attn, const float* __restrict__ values,
    float* __restrict__ out) {
  __shared__ float s_b[2][KT_ * 16];      // 2 x 4KB double buffer

  const int tid    = threadIdx.x;
  const int lane   = tid & 31;
  const int waveId = tid >> 5;            // 0..3 -> m0 = waveId*64
  const int b  = blockIdx.x / (H_ / 16);
  const int n0 = (blockIdx.x % (H_ / 16)) * 16;
  const int m0 = waveId * 64;

  // staging map: thread t -> row = t/4 (+32 for 2nd issue), 4 floats at col 4*(t%4)
  const int srow = tid >> 2;              // 0..31
  const int scol = (tid & 3) << 2;        // 0,4,8,12
  const float* gsrc = values + (size_t)b * V_ * H_ + (size_t)srow * H_ + n0 + scol;

  const int mi = lane & 15;               // A: lane -> M row
  const int kk = (lane >> 4) << 1;        // upper half-wave -> K+2
  const int nn = lane & 15;               // B: lane -> N col
  const float* arow = attn + ((size_t)b * Q_ + m0 + mi) * V_;

  v8f acc0 = {}, acc1 = {}, acc2 = {}, acc3 = {};

  // prologue: stage chunk 0 into buffer 0
  async_copy16(gsrc, &s_b[0][srow * 16 + scol]);
  async_copy16(gsrc + (size_t)32 * H_, &s_b[0][(srow + 32) * 16 + scol]);

  const int NC = V_ / KT_;                // 32 chunks
  for (int c = 0; c < NC; ++c) {
    const int k0 = c * KT_;
    if (c + 1 < NC) {                     // prefetch next chunk into other buffer
      const float* g = gsrc + (size_t)(k0 + KT_) * H_;
      float* l = &s_b[(c + 1) & 1][0];
      async_copy16(g, l + srow * 16 + scol);
      async_copy16(g + (size_t)32 * H_, l + (srow + 32) * 16 + scol);
      wait_async2();                      // current chunk's 2 copies retired
    } else {
      wait_async0();
    }
    __syncthreads();                      // all waves' staging visible

    const float* sb = &s_b[c & 1][0];
    for (int k = 0; k < KT_; k += 4) {
      v2f bfrag;
      bfrag.x = sb[(k + kk) * 16 + nn];
      bfrag.y = sb[(k + kk + 1) * 16 + nn];
      v2f a_0 = *(const v2f*)(arow + 0 * 16 * V_ + k0 + k + kk);
      v2f a_1 = *(const v2f*)(arow + 1 * 16 * V_ + k0 + k + kk);
      v2f a_2 = *(const v2f*)(arow + 2 * 16 * V_ + k0 + k + kk);
      v2f a_3 = *(const v2f*)(arow + 3 * 16 * V_ + k0 + k + kk);
      acc0 = __builtin_amdgcn_wmma_f32_16x16x4_f32(false, a_0, false, bfrag, (short)0, acc0, false, false);
      acc1 = __builtin_amdgcn_wmma_f32_16x16x4_f32(false, a_1, false, bfrag, (short)0, acc1, false, false);
      acc2 = __builtin_amdgcn_wmma_f32_16x16x4_f32(false, a_2, false, bfrag, (short)0, acc2, false, false);
      acc3 = __builtin_amdgcn_wmma_f32_16x16x4_f32(false, a_3, false, bfrag, (short)0, acc3, false, false);
    }
    __syncthreads();                      // done reading before next overwrite
  }

  const int mo = (lane >> 4) * 8;         // C/D: upper half-wave -> M+8
  const int n  = n0 + (lane & 15);
  for (int r = 0; r < 8; ++r) {
    out[((size_t)b * Q_ + m0 + 0 * 16 + mo + r) * H_ + n] = acc0[r];
    out[((size_t)b * Q_ + m0 + 1 * 16 + mo + r) * H_ + n] = acc1[r];
    out[((size_t)b * Q_ + m0 + 2 * 16 + mo + r) * H_ + n] = acc2[r];
    out[((size_t)b * Q_ + m0 + 3 * 16 + mo + r) * H_ + n] = acc3[r];
  }
}

extern "C" void kernel_launch(void* const* d_in, const int* in_sizes, int n_in,
                              void* d_out, int out_size, void* d_ws, size_t ws_size,
                              hipStream_t stream) {
  (void)in_sizes; (void)n_in; (void)out_size; (void)ws_size;
  const float* queries = (const float*)d_in[0];  // [B,Q,H]
  const float* values  = (const float*)d_in[1];  // [B,V,H]
  const float* w1      = (const float*)d_in[2];  // [H,U]
  const float* w2      = (const float*)d_in[3];  // [H,U]
  const float* vvec    = (const float*)d_in[4];  // [U]
  float* out = (float*)d_out;                    // [B,Q,H]

  // Workspace layout (floats): pq 512KB | pv 4MB | attn 8MB  (~12.5MB total)
  float* pq   = (float*)d_ws;                    // [B*Q, U]
  float* pv   = pq + (size_t)B_ * Q_ * U_;       // [B*V, U]
  float* attn = pv + (size_t)B_ * V_ * U_;       // [B*Q, V]

  // K1: (B*(Q+V))/64 * (U/16) waves = 1152 waves -> 288 blocks of 4 waves
  proj_kernel<<<288, 128, 0, stream>>>(queries, values, w1, w2, pq, pv);
  // K2: one workgroup per (b,q)
  score_softmax_kernel<<<B_ * Q_, 256, 0, stream>>>(pq, pv, vvec, attn);
  // K3: one block per (b, 16-col strip of H): 4*32 = 128 blocks
  out_gemm_kernel<<<B_ * (H_ / 16), 128, 0, stream>>>(attn, values, out);
}